// SelfAttention_16578573762801
// MI455X (gfx1250) — compile-verified
//
#include <hip/hip_runtime.h>
#include <hip/hip_bf16.h>

// ---------------------------------------------------------------------------
// Self-attention for MI455X (gfx1250), fp32 end-to-end using
// V_WMMA_F32_16X16X4_F32 + GLOBAL_LOAD_ASYNC_TO_LDS_B128 staging.
//
// Shapes: B=8, T=1024, C=1024, H=16, D=64.
//   1) qkv = x @ w_qkv            (8192x1024 @ 1024x3072)  -> ws [3][B][H][T][D]
//   2) 2D RoPE on q,k (elementwise)
//   3) flash attention per (b,h,64-row q block)            -> ws attn [B*T][1024]
//   4) y = attn @ w_proj          (8192x1024 @ 1024x1024)  -> d_out
// ---------------------------------------------------------------------------

typedef __attribute__((ext_vector_type(2))) float v2f;
typedef __attribute__((ext_vector_type(8))) float v8f;

#define N_DIM   1024
#define N_HEAD  16
#define HEAD_D  64
#define SEQ_T   1024
#define BATCH   8
#define BHTD    (BATCH * N_HEAD * SEQ_T * HEAD_D)   // 8,388,608 elements / matrix

__device__ __forceinline__ v8f wmma4(v2f a, v2f b, v8f c) {
  // D = A(16x4,f32) * B(4x16,f32) + C(16x16,f32)
  return __builtin_amdgcn_wmma_f32_16x16x4_f32(
      /*neg_a=*/false, a, /*neg_b=*/false, b,
      /*c_mod=*/(short)0, c, /*reuse_a=*/false, /*reuse_b=*/false);
}

// Async memory->LDS copy, 16 bytes per lane, tracked by ASYNCcnt.
__device__ __forceinline__ void async_copy16(void* lds_dst, const void* gsrc) {
  unsigned lds_off = (unsigned)(size_t)lds_dst;   // low 32 bits = LDS byte offset
  asm volatile("global_load_async_to_lds_b128 %0, %1, off"
               :: "v"(lds_off), "v"(gsrc) : "memory");
}
__device__ __forceinline__ void wait_async0() {
  asm volatile("s_wait_asynccnt 0" ::: "memory");
}

// ---------------------------------------------------------------------------
// GEMM: out(MxN) = A(MxK) * W(KxN), fp32, WMMA f32 16x16x4.
// Block tile 128x64, 8 waves, each wave 32x32 (2x2 C tiles).
// Double-buffered LDS, async global->LDS staging.
// mode 0: out[row*N + col] (proj)
// mode 1: scatter qkv into [which][B][H][T][D]
// ---------------------------------------------------------------------------
__global__ __launch_bounds__(256) void gemm_kernel(
    const float* __restrict__ A, const float* __restrict__ W,
    float* __restrict__ out, int M, int N, int K, int mode) {
  const int tid  = threadIdx.x;
  const int m0   = blockIdx.y * 128;
  const int n0   = blockIdx.x * 64;
  const int w    = tid >> 5;
  const int lane = tid & 31;
  const int lrow = lane & 15;
  const int lhi  = lane >> 4;            // 0: K={0,1}, 1: K={2,3}
  const int mo   = (w >> 1) * 32;        // wave M offset in block tile
  const int no   = (w & 1) * 32;         // wave N offset in block tile

  // stride 20 floats: 16B-aligned rows, 16 distinct banks for column reads
  __shared__ float lds_a[2][128][20];
  __shared__ float lds_b[2][16][64];

  // staging coordinates (fixed per thread)
  const int ar0 = (tid)       >> 2, ac0 = (tid)       & 3;
  const int ar1 = (tid + 256) >> 2, ac1 = (tid + 256) & 3;
  const int br  = tid >> 4,         bc  = tid & 15;

  v8f c00 = {}, c01 = {}, c10 = {}, c11 = {};

  // prologue: stage first K-chunk into buffer 0
  async_copy16(&lds_a[0][ar0][ac0 * 4], A + (size_t)(m0 + ar0) * K + ac0 * 4);
  async_copy16(&lds_a[0][ar1][ac1 * 4], A + (size_t)(m0 + ar1) * K + ac1 * 4);
  async_copy16(&lds_b[0][br][bc * 4],   W + (size_t)br * N + n0 + bc * 4);

  for (int k0 = 0; k0 < K; k0 += 16) {
    const int buf = (k0 >> 4) & 1;
    wait_async0();         // this wave's async writes visible
    __syncthreads();       // all waves' writes visible, prior compute done

    if (k0 + 16 < K) {     // overlap: stage next chunk while computing this one
      const int nb = buf ^ 1, kn = k0 + 16;
      async_copy16(&lds_a[nb][ar0][ac0 * 4], A + (size_t)(m0 + ar0) * K + kn + ac0 * 4);
      async_copy16(&lds_a[nb][ar1][ac1 * 4], A + (size_t)(m0 + ar1) * K + kn + ac1 * 4);
      async_copy16(&lds_b[nb][br][bc * 4],   W + (size_t)(kn + br) * N + n0 + bc * 4);
    }

#pragma unroll
    for (int kk = 0; kk < 16; kk += 4) {
      int ka = kk + lhi * 2;
      v2f a0 = {lds_a[buf][mo +      lrow][ka], lds_a[buf][mo +      lrow][ka + 1]};
      v2f a1 = {lds_a[buf][mo + 16 + lrow][ka], lds_a[buf][mo + 16 + lrow][ka + 1]};
      v2f b0 = {lds_b[buf][ka][no +      lrow], lds_b[buf][ka + 1][no +      lrow]};
      v2f b1 = {lds_b[buf][ka][no + 16 + lrow], lds_b[buf][ka + 1][no + 16 + lrow]};
      c00 = wmma4(a0, b0, c00);
      c01 = wmma4(a0, b1, c01);
      c10 = wmma4(a1, b0, c10);
      c11 = wmma4(a1, b1, c11);
    }
  }

  // ---- epilogue ----
#pragma unroll
  for (int ti = 0; ti < 2; ++ti) {
#pragma unroll
    for (int tj = 0; tj < 2; ++tj) {
      v8f cc = (ti == 0) ? (tj == 0 ? c00 : c01) : (tj == 0 ? c10 : c11);
#pragma unroll
      for (int j = 0; j < 8; ++j) {
        int gm = m0 + mo + ti * 16 + j + lhi * 8;
        int gn = n0 + no + tj * 16 + lrow;
        float val = cc[j];
        if (mode == 0) {
          out[(size_t)gm * N + gn] = val;
        } else {
          int which = gn >> 10;              // 0=q,1=k,2=v
          int c     = gn & 1023;
          int hh    = c >> 6, d = c & 63;
          int bi    = gm >> 10, t = gm & 1023;
          size_t addr = (size_t)which * BHTD +
                        (((size_t)(bi * N_HEAD + hh) << 10) + t) * HEAD_D + d;
          out[addr] = val;
        }
      }
    }
  }
}

// ---------------------------------------------------------------------------
// 2D RoPE on q and k, layout [which(2)][B][H][T][D]. One thread per pair.
// d in [0,32): axis 0 (pos[...,0]); d in [32,64): axis 1 (pos[...,1]).
// theta_i = 10000^(-i/16), i = pair index in [0,16).
// ---------------------------------------------------------------------------
__global__ __launch_bounds__(256) void rope_kernel(float* __restrict__ qk,
                                                   const int* __restrict__ pos) {
  int idx = blockIdx.x * 256 + threadIdx.x;   // 2*8*16*1024*32 = 8,388,608
  int d2   = idx & 31;  int rest = idx >> 5;
  int t    = rest & 1023; rest >>= 10;
  int h    = rest & 15;   rest >>= 4;
  int bb   = rest & 7;    int which = rest >> 3;
  int axis = d2 >> 4;     int i = d2 & 15;

  int p = pos[((bb << 10) + t) * 2 + axis];
  float theta = __expf(-9.210340372f * (float)i * (1.0f / 16.0f)); // 10000^(-i/16)
  int pc = p > 0 ? p : 0;
  float ang = (float)pc * theta;
  float sn, cs;
  sincosf(ang, &sn, &cs);

  size_t base = (size_t)which * BHTD +
                (((size_t)(bb * N_HEAD + h) << 10) + t) * HEAD_D +
                (axis * 32 + 2 * i);
  float x0 = qk[base], x1 = qk[base + 1];
  if (p >= 0) {
    qk[base]     = x0 * cs - x1 * sn;
    qk[base + 1] = x1 * cs + x0 * sn;
  }
}

// ---------------------------------------------------------------------------
// Flash attention per (b, h, 64-query block); 32-key K/V blocks streamed.
// 8 waves: S tile = 16x16 per wave (64x32 total); O = 16x32 per wave.
// Async global->LDS staging for Q/K/V tiles.
// Output written as [B*T][H*D] row-major (ready as GEMM A for proj).
// ---------------------------------------------------------------------------
__global__ __launch_bounds__(256) void attn_kernel(
    const float* __restrict__ Q, const float* __restrict__ Kc,
    const float* __restrict__ V, const unsigned char* __restrict__ maskp,
    float* __restrict__ out) {
  const int tid = threadIdx.x;
  const int b = blockIdx.z, h = blockIdx.y, q0 = blockIdx.x * 64;

  // stride 68 floats: 16B-aligned rows, banks 4 apart for column reads
  __shared__ float lds_q[64][68];
  __shared__ float lds_k[32][68];
  __shared__ float lds_v[32][68];
  __shared__ float lds_p[64][33];
  __shared__ float m_run[64], l_run[64], fac[64];

  const size_t headbase = ((size_t)(b * N_HEAD + h) << 10) * HEAD_D;

  // async-load Q block 64x64 (4 x 16B chunks per thread)
#pragma unroll
  for (int i = 0; i < 4; ++i) {
    int e = tid + i * 256;
    int row = e >> 4, c4 = e & 15;
    async_copy16(&lds_q[row][c4 * 4],
                 Q + headbase + (size_t)(q0 + row) * HEAD_D + c4 * 4);
  }
  if (tid < 64) { m_run[tid] = -3.0e38f; l_run[tid] = 0.0f; }

  const int w = tid >> 5, lane = tid & 31;
  const int lrow = lane & 15, lhi = lane >> 4;
  const int sm0 = (w >> 1) * 16, sn0 = (w & 1) * 16;   // S tile offsets
  const int om0 = (w >> 1) * 16, on0 = (w & 1) * 32;   // O tile offsets
  const int kvr0 = tid >> 4,        kvc0 = tid & 15;   // K/V staging coords
  const int kvr1 = (tid + 256) >> 4, kvc1 = (tid + 256) & 15;
  const float scale = 0.125f;                          // 1/sqrt(64)

  v8f o0 = {}, o1 = {};

  for (int kb = 0; kb < SEQ_T / 32; ++kb) {
    const int kg = kb * 32;
    // ---- async-stage K,V blocks 32x64 (prev iter's reads fenced by loop-end barrier)
    async_copy16(&lds_k[kvr0][kvc0 * 4],
                 Kc + headbase + (size_t)(kg + kvr0) * HEAD_D + kvc0 * 4);
    async_copy16(&lds_k[kvr1][kvc1 * 4],
                 Kc + headbase + (size_t)(kg + kvr1) * HEAD_D + kvc1 * 4);
    async_copy16(&lds_v[kvr0][kvc0 * 4],
                 V + headbase + (size_t)(kg + kvr0) * HEAD_D + kvc0 * 4);
    async_copy16(&lds_v[kvr1][kvc1 * 4],
                 V + headbase + (size_t)(kg + kvr1) * HEAD_D + kvc1 * 4);
    wait_async0();
    __syncthreads();

    // ---- S = Q @ K^T (per-wave 16x16), reduce over d=64 ----
    v8f s = {};
#pragma unroll
    for (int kk = 0; kk < HEAD_D; kk += 4) {
      int ka = kk + lhi * 2;
      v2f a  = {lds_q[sm0 + lrow][ka], lds_q[sm0 + lrow][ka + 1]};
      v2f bt = {lds_k[sn0 + lrow][ka], lds_k[sn0 + lrow][ka + 1]};  // K^T frag
      s = wmma4(a, bt, s);
    }
#pragma unroll
    for (int j = 0; j < 8; ++j) {
      int r  = sm0 + j + lhi * 8;
      int cl = sn0 + lrow;
      float val = s[j] * scale;
      if (!maskp[b * SEQ_T + kg + cl]) val = -1.0e30f;
      lds_p[r][cl] = val;
    }
    __syncthreads();

    // ---- online softmax, one row per thread (64 rows) ----
    if (tid < 64) {
      int r = tid;
      float mo = m_run[r], mx = mo;
#pragma unroll
      for (int j = 0; j < 32; ++j) mx = fmaxf(mx, lds_p[r][j]);
      float f = __expf(mo - mx);
      float sum = 0.0f;
#pragma unroll
      for (int j = 0; j < 32; ++j) {
        float p = __expf(lds_p[r][j] - mx);
        lds_p[r][j] = p;
        sum += p;
      }
      l_run[r] = l_run[r] * f + sum;
      m_run[r] = mx;
      fac[r]   = f;
    }
    __syncthreads();

    // ---- rescale O accumulators, then O += P @ V ----
#pragma unroll
    for (int j = 0; j < 8; ++j) {
      float f = fac[om0 + j + lhi * 8];
      o0[j] *= f;
      o1[j] *= f;
    }
#pragma unroll
    for (int kk = 0; kk < 32; kk += 4) {
      int ka = kk + lhi * 2;
      v2f a  = {lds_p[om0 + lrow][ka], lds_p[om0 + lrow][ka + 1]};
      v2f b0 = {lds_v[ka][on0 +      lrow], lds_v[ka + 1][on0 +      lrow]};
      v2f b1 = {lds_v[ka][on0 + 16 + lrow], lds_v[ka + 1][on0 + 16 + lrow]};
      o0 = wmma4(a, b0, o0);
      o1 = wmma4(a, b1, o1);
    }
    __syncthreads();   // fence lds_k/v/p reads before next block's async writes
  }

  // ---- epilogue: normalize and write [B*T][H*D] ----
#pragma unroll
  for (int j = 0; j < 8; ++j) {
    int r = om0 + j + lhi * 8;
    float inv = 1.0f / l_run[r];
    int t = q0 + r;
    size_t ob = ((size_t)(b * SEQ_T + t)) * N_DIM + h * HEAD_D;
    out[ob + on0 + lrow]      = o0[j] * inv;
    out[ob + on0 + 16 + lrow] = o1[j] * inv;
  }
}

// ---------------------------------------------------------------------------
extern "C" void kernel_launch(void* const* d_in, const int* in_sizes, int n_in,
                              void* d_out, int out_size, void* d_ws, size_t ws_size,
                              hipStream_t stream) {
  const float*         x         = (const float*)d_in[0];
  const unsigned char* attn_mask = (const unsigned char*)d_in[1];
  const int*           positions = (const int*)d_in[2];
  const float*         w_qkv     = (const float*)d_in[3];
  const float*         w_proj    = (const float*)d_in[4];

  float* qkv  = (float*)d_ws;            // [3][B][H][T][D] = 96 MB
  float* attn = qkv + (size_t)3 * BHTD;  // [B*T][1024]     = 32 MB
  float* q = qkv;
  float* k = qkv + (size_t)BHTD;
  float* v = qkv + (size_t)2 * BHTD;

  // 1) qkv = x @ w_qkv, scattered to [3][B][H][T][D]
  gemm_kernel<<<dim3(3 * N_DIM / 64, BATCH * SEQ_T / 128), 256, 0, stream>>>(
      x, w_qkv, qkv, BATCH * SEQ_T, 3 * N_DIM, N_DIM, /*mode=*/1);

  // 2) RoPE on q and k (contiguous in ws)
  rope_kernel<<<(2 * BHTD / 2) / 256, 256, 0, stream>>>(qkv, positions);

  // 3) flash attention -> attn [B*T][1024]
  attn_kernel<<<dim3(SEQ_T / 64, N_HEAD, BATCH), 256, 0, stream>>>(
      q, k, v, attn_mask, attn);

  // 4) y = attn @ w_proj -> d_out
  gemm_kernel<<<dim3(N_DIM / 64, BATCH * SEQ_T / 128), 256, 0, stream>>>(
      attn, w_proj, (float*)d_out, BATCH * SEQ_T, N_DIM, N_DIM, /*mode=*/0);
}